// NAFBlockModulated_87763361727185
// MI455X (gfx1250) — compile-verified
//
#include <hip/hip_runtime.h>
#include <hip/hip_bf16.h>

// Geometry constants from the reference
#define NB   8      // batch
#define NC   32     // channels C
#define NDW  64     // DW
#define NW   8      // WH == WW (windows per side)
#define PH   32     // pixels per window side (256/8)
#define NPIX 1024   // 32*32 pixels per window
#define RING 34     // 32 + 2 halo
#define RPIX (RING*RING)   // 1156
#define IMGW 256
#define IMG2 65536  // 256*256

typedef __attribute__((ext_vector_type(2))) float v2f;
typedef __attribute__((ext_vector_type(8))) float v8f;

static __device__ __forceinline__ v8f wmma4(v2f a, v2f b, v8f c) {
  // D(16x16,f32) = A(16x4,f32) * B(4x16,f32) + C
  return __builtin_amdgcn_wmma_f32_16x16x4_f32(
      /*neg_a=*/false, a, /*neg_b=*/false, b,
      /*c_mod=*/(short)0, c, /*reuse_a=*/false, /*reuse_b=*/false);
}

// ---------------------------------------------------------------------------
// K1: LN1 + modulated 1x1 (32->64, WMMA) + modulated depthwise 3x3 + gate.
// One block per (batch, window). Writes gated y1 and per-window pool partials.
// ---------------------------------------------------------------------------
__global__ __launch_bounds__(256) void k1_ln_conv1_dw3_gate(
    const float* __restrict__ x, const float* __restrict__ mod1,
    const float* __restrict__ mod2,
    const float* __restrict__ w1, const float* __restrict__ b1,
    const float* __restrict__ w2, const float* __restrict__ b2,
    const float* __restrict__ ln1g, const float* __restrict__ ln1b,
    float* __restrict__ y1buf, float* __restrict__ ppbuf)
{
  extern __shared__ float sm[];
  float* XN  = sm;                  // 32*1156  LN'd input, 34x34 tile
  float* Y0  = XN  + NC * RPIX;     // 32*1156  conv1 out (current ch group)
  float* WN1 = Y0  + NC * RPIX;     // 32*32    normalized conv1 rows (own win)
  float* WNE = WN1 + 1024;          // 32*32    normalized rows (edge neighbor)
  float* W2N = WNE + 1024;          // 32*12    normalized 3x3 taps
  float* B1V = W2N + 384;           // 32
  float* B2V = B1V + 32;            // 32
  float* LG  = B2V + 32;            // 32
  float* LB  = LG  + 32;            // 32
  float* RED = LB  + 32;            // 256

  const int t    = threadIdx.x;
  const int lane = t & 31;
  const int wv   = t >> 5;                  // wave 0..7
  const int blk  = blockIdx.x;              // b*64 + widx
  const int b    = blk >> 6;
  const int widx = blk & 63;
  const int wi   = widx >> 3, wj = widx & 7;

  if (t < 32) { LG[t] = ln1g[t]; LB[t] = ln1b[t]; }
  __syncthreads();

  // ---- load x tile (34x34 with halo), per-pixel layernorm over channels ----
  for (int pix = t; pix < RPIX; pix += 256) {
    const int py = pix / RING - 1, px = pix % RING - 1;
    const int gy = wi * PH + py, gx = wj * PH + px;
    const bool in = (gy >= 0) && (gy < IMGW) && (gx >= 0) && (gx < IMGW);
    float s1 = 0.f, s2 = 0.f;
    const long gb = ((long)(b * NC) * IMGW + gy) * IMGW + gx;
    for (int c = 0; c < NC; ++c) {
      float v = in ? x[gb + (long)c * IMG2] : 0.f;
      XN[c * RPIX + pix] = v; s1 += v; s2 += v * v;
    }
    const float mu = s1 * (1.f / 32.f);
    const float rs = rsqrtf(s2 * (1.f / 32.f) - mu * mu + 1e-6f);
    for (int c = 0; c < NC; ++c) {
      float v = XN[c * RPIX + pix];
      XN[c * RPIX + pix] = (v - mu) * rs * LG[c] + LB[c];
    }
  }
  __syncthreads();

  // ---- two channel-group passes: g=0 -> out ch {0..15,32..47}; g=1 -> rest
  for (int g = 0; g < 2; ++g) {
    // --- normalize modulated weights for this window / group ---
    if (t < 32) {
      const int o = (t < 16) ? (g * 16 + t) : (32 + g * 16 + (t - 16));
      float ss = 0.f;
      for (int c = 0; c < NC; ++c) {
        float wvv = w1[o * NC + c] *
                    mod1[((long)(b * NDW + o) * NC + c) * 64 + widx];
        WN1[t * 32 + c] = wvv; ss += wvv * wvv;
      }
      const float rn = rsqrtf(ss);
      for (int c = 0; c < NC; ++c) WN1[t * 32 + c] *= rn;
      B1V[t] = b1[o];
      float s9 = 0.f;
      for (int j = 0; j < 9; ++j) {
        float wvv = w2[o * 9 + j] * mod2[((long)(b * NDW + o) * 9 + j) * 64 + widx];
        W2N[t * 12 + j] = wvv; s9 += wvv * wvv;
      }
      const float r9 = rsqrtf(s9);
      for (int j = 0; j < 9; ++j) W2N[t * 12 + j] *= r9;
      B2V[t] = b2[o];
    }
    __syncthreads();

    // --- interior conv1 as WMMA GEMM: 32 out-slots x 1024 px, K=32 ---
    {
      const int l15   = lane & 15;
      const int khalf = (lane >> 4) * 2;
      const int mhi   = (lane >> 4) * 8;
      for (int nt = wv * 8; nt < wv * 8 + 8; ++nt) {
        const int n  = nt * 16 + l15;
        const int py = n >> 5, px = n & 31;
        const int bidx = (py + 1) * RING + (px + 1);
        v8f d0 = {0.f,0.f,0.f,0.f,0.f,0.f,0.f,0.f};
        v8f d1 = d0;
#pragma unroll
        for (int kc = 0; kc < 8; ++kc) {
          const int c0 = kc * 4 + khalf;
          v2f a0, a1, bb;
          a0.x = WN1[l15 * 32 + c0];        a0.y = WN1[l15 * 32 + c0 + 1];
          a1.x = WN1[(16 + l15) * 32 + c0]; a1.y = WN1[(16 + l15) * 32 + c0 + 1];
          bb.x = XN[c0 * RPIX + bidx];      bb.y = XN[(c0 + 1) * RPIX + bidx];
          d0 = wmma4(a0, bb, d0);
          d1 = wmma4(a1, bb, d1);
        }
#pragma unroll
        for (int vvv = 0; vvv < 8; ++vvv) {
          const int m = vvv + mhi;
          Y0[m * RPIX + bidx]        = d0[vvv] + B1V[m];
          Y0[(16 + m) * RPIX + bidx] = d1[vvv] + B1V[16 + m];
        }
      }
    }
    __syncthreads();

    // --- halo: conv1 with the NEIGHBOR window's normalized weights ---
    for (int e = 0; e < 4; ++e) {
      const int nwi = wi + ((e == 0) ? -1 : (e == 1) ? 1 : 0);
      const int nwj = wj + ((e == 2) ? -1 : (e == 3) ? 1 : 0);
      const bool val = (nwi >= 0) && (nwi < NW) && (nwj >= 0) && (nwj < NW);
      if (val && t < 32) {
        const int o = (t < 16) ? (g * 16 + t) : (32 + g * 16 + (t - 16));
        const int nix = nwi * 8 + nwj;
        float ss = 0.f;
        for (int c = 0; c < NC; ++c) {
          float wvv = w1[o * NC + c] *
                      mod1[((long)(b * NDW + o) * NC + c) * 64 + nix];
          WNE[t * 32 + c] = wvv; ss += wvv * wvv;
        }
        const float rn = rsqrtf(ss);
        for (int c = 0; c < NC; ++c) WNE[t * 32 + c] *= rn;
      }
      __syncthreads();
      for (int it = 0; it < 4; ++it) {
        const int idx = t + it * 256;       // 32 px * 32 slots
        const int slot = idx >> 5, j = idx & 31;
        int ppy, ppx;
        if (e == 0)      { ppy = -1; ppx = j;  }
        else if (e == 1) { ppy = PH; ppx = j;  }
        else if (e == 2) { ppy = j;  ppx = -1; }
        else             { ppy = j;  ppx = PH; }
        const int pidx = (ppy + 1) * RING + (ppx + 1);
        float r = 0.f;
        if (val) {
          float acc = 0.f;
          for (int c = 0; c < NC; ++c) acc += WNE[slot * 32 + c] * XN[c * RPIX + pidx];
          r = acc + B1V[slot];
        }
        Y0[slot * RPIX + pidx] = r;
      }
      __syncthreads();
    }
    // corners: on-the-fly neighbor normalization (4 px * 32 slots)
    if (t < 128) {
      const int corner = t >> 5, slot = t & 31;
      const int cpy = (corner < 2) ? -1 : PH;
      const int cpx = (corner & 1) ? PH : -1;
      const int nwi = wi + ((corner < 2) ? -1 : 1);
      const int nwj = wj + ((corner & 1) ? 1 : -1);
      const int pidx = (cpy + 1) * RING + (cpx + 1);
      float r = 0.f;
      if ((nwi >= 0) && (nwi < NW) && (nwj >= 0) && (nwj < NW)) {
        const int o = (slot < 16) ? (g * 16 + slot) : (32 + g * 16 + (slot - 16));
        const int nix = nwi * 8 + nwj;
        float ss = 0.f, acc = 0.f;
        for (int c = 0; c < NC; ++c) {
          float wvv = w1[o * NC + c] *
                      mod1[((long)(b * NDW + o) * NC + c) * 64 + nix];
          ss += wvv * wvv;
          acc += wvv * XN[c * RPIX + pidx];
        }
        r = acc * rsqrtf(ss) + B1V[slot];
      }
      Y0[slot * RPIX + pidx] = r;
    }
    __syncthreads();

    // --- depthwise 3x3 (per-window taps) + gate + pool partials ---
    {
      const int kslot = t >> 4;          // 0..15 -> gated channel in group
      const int lg16  = t & 15;
      const int chg   = g * 16 + kslot;
      float lsum = 0.f;
      const float biasA = B2V[kslot], biasB = B2V[16 + kslot];
      const long obase = ((long)(b * NC + chg)) << 16;
      for (int i = 0; i < 64; ++i) {
        const int p  = lg16 + (i << 4);
        const int py = p >> 5, px = p & 31;
        const int c0 = py * RING + px;   // top-left of 3x3 stencil in Y0
        float a3 = biasA, b3v = biasB;
#pragma unroll
        for (int dh = 0; dh < 3; ++dh)
#pragma unroll
          for (int dw_ = 0; dw_ < 3; ++dw_) {
            const int o2 = c0 + dh * RING + dw_;
            a3  += W2N[kslot * 12 + dh * 3 + dw_]        * Y0[kslot * RPIX + o2];
            b3v += W2N[(16 + kslot) * 12 + dh * 3 + dw_] * Y0[(16 + kslot) * RPIX + o2];
          }
        const float gv = a3 * b3v;
        y1buf[obase + (long)(wi * PH + py) * IMGW + (wj * PH + px)] = gv;
        lsum += gv;
      }
      RED[t] = lsum;
    }
    __syncthreads();
    if ((t & 15) == 0) {
      float s = 0.f;
      for (int j = 0; j < 16; ++j) s += RED[t + j];
      const int chg = g * 16 + (t >> 4);
      ppbuf[(b * NC + chg) * 64 + widx] = s;   // deterministic partials
    }
    __syncthreads();
  }
}

// ---------------------------------------------------------------------------
// K2: reduce pool partials -> pooled mean; att = sca_w @ pooled + sca_b
// ---------------------------------------------------------------------------
__global__ __launch_bounds__(256) void k2_sca(
    const float* __restrict__ pp, const float* __restrict__ scaw,
    const float* __restrict__ scab, float* __restrict__ attb)
{
  __shared__ float PL[256];
  const int t = threadIdx.x;
  const int b = t >> 5, c = t & 31;
  float s = 0.f;
  for (int w = 0; w < 64; ++w) s += pp[(b * NC + c) * 64 + w];
  PL[t] = s * (1.f / 65536.f);
  __syncthreads();
  float a = scab[c];
  for (int k = 0; k < NC; ++k) a += scaw[c * NC + k] * PL[b * NC + k];
  attb[b * NC + c] = a;
}

// ---------------------------------------------------------------------------
// K3: y1*att -> 1x1(w3,WMMA) -> z = x + .*beta (LDS-resident) -> LN2
//     -> 1x1(w4,WMMA, 4 M-tiles) -> gate on D-fragments -> 1x1(w5,WMMA)
//     -> out = z + .*gamma.  Fully column-local per wave (128 px each).
// ---------------------------------------------------------------------------
__global__ __launch_bounds__(256) void k3_sca_apply_ffn(
    const float* __restrict__ x, const float* __restrict__ y1buf,
    const float* __restrict__ attb,
    const float* __restrict__ mod3, const float* __restrict__ mod4,
    const float* __restrict__ mod5,
    const float* __restrict__ w3, const float* __restrict__ b3,
    const float* __restrict__ w4, const float* __restrict__ b4,
    const float* __restrict__ w5, const float* __restrict__ b5,
    const float* __restrict__ ln2g, const float* __restrict__ ln2b,
    const float* __restrict__ beta, const float* __restrict__ gamma,
    float* __restrict__ outp)
{
  extern __shared__ float sm[];
  float* BUFA = sm;                 // 32*1024 (y1*att -> xn2 -> gated h)
  float* BUFB = BUFA + NC * NPIX;   // 32*1024 (z, kept resident)
  float* W3N  = BUFB + NC * NPIX;   // 32*32
  float* W4N  = W3N + 1024;         // 64*32
  float* W5N  = W4N + 2048;         // 32*32
  float* B3V  = W5N + 1024;         // 32
  float* B4V  = B3V + 32;           // 64
  float* B5V  = B4V + 64;           // 32
  float* LG2  = B5V + 32;           // 32
  float* LB2  = LG2 + 32;           // 32
  float* BTA  = LB2 + 32;           // 32
  float* GMA  = BTA + 32;           // 32
  float* ATT  = GMA + 32;           // 32

  const int t    = threadIdx.x;
  const int lane = t & 31;
  const int wvid = t >> 5;
  const int blk  = blockIdx.x;
  const int b    = blk >> 6;
  const int widx = blk & 63;
  const int wi   = widx >> 3, wj = widx & 7;

  // --- normalize modulated weights for this window ---
  if (t < 32) {
    float ss = 0.f;
    for (int c = 0; c < 32; ++c) {
      float w_ = w3[t * 32 + c] * mod3[((long)(b * NC + t) * 32 + c) * 64 + widx];
      W3N[t * 32 + c] = w_; ss += w_ * w_;
    }
    float rn = rsqrtf(ss);
    for (int c = 0; c < 32; ++c) W3N[t * 32 + c] *= rn;
    B3V[t] = b3[t];
    float s5 = 0.f;
    for (int c = 0; c < 32; ++c) {
      float w_ = w5[t * 32 + c] * mod5[((long)(b * NC + t) * 32 + c) * 64 + widx];
      W5N[t * 32 + c] = w_; s5 += w_ * w_;
    }
    float r5 = rsqrtf(s5);
    for (int c = 0; c < 32; ++c) W5N[t * 32 + c] *= r5;
    B5V[t] = b5[t];
    LG2[t] = ln2g[t]; LB2[t] = ln2b[t];
    BTA[t] = beta[t]; GMA[t] = gamma[t];
    ATT[t] = attb[b * NC + t];
  }
  if (t < 64) {
    float ss = 0.f;
    for (int c = 0; c < 32; ++c) {
      float w_ = w4[t * 32 + c] * mod4[((long)(b * NDW + t) * 32 + c) * 64 + widx];
      W4N[t * 32 + c] = w_; ss += w_ * w_;
    }
    float rn = rsqrtf(ss);
    for (int c = 0; c < 32; ++c) W4N[t * 32 + c] *= rn;
    B4V[t] = b4[t];
  }
  __syncthreads();

  const int pbase = wvid * 128;     // this wave owns 128 pixel columns
  // --- stage y1 * att ---
  for (int c = 0; c < 32; ++c) {
    const float av = ATT[c];
    const long gb = ((long)(b * NC + c)) << 16;
    for (int i = 0; i < 4; ++i) {
      const int p = pbase + i * 32 + lane;
      const int py = p >> 5, px = p & 31;
      BUFA[c * NPIX + p] =
          y1buf[gb + (long)(wi * PH + py) * IMGW + (wj * PH + px)] * av;
    }
  }
  __syncthreads();

  const int l15   = lane & 15;
  const int khalf = (lane >> 4) * 2;
  const int mhi   = (lane >> 4) * 8;

  // --- GEMM1 (w3) ; z = x + (u+b3)*beta -> BUFB ---
  for (int nt = wvid * 8; nt < wvid * 8 + 8; ++nt) {
    const int n = nt * 16 + l15;
    const int py = n >> 5, px = n & 31;
    v8f d0 = {0.f,0.f,0.f,0.f,0.f,0.f,0.f,0.f};
    v8f d1 = d0;
#pragma unroll
    for (int kc = 0; kc < 8; ++kc) {
      const int c0 = kc * 4 + khalf;
      v2f a0, a1, bb;
      a0.x = W3N[l15 * 32 + c0];        a0.y = W3N[l15 * 32 + c0 + 1];
      a1.x = W3N[(16 + l15) * 32 + c0]; a1.y = W3N[(16 + l15) * 32 + c0 + 1];
      bb.x = BUFA[c0 * NPIX + n];       bb.y = BUFA[(c0 + 1) * NPIX + n];
      d0 = wmma4(a0, bb, d0);
      d1 = wmma4(a1, bb, d1);
    }
    const long gpix = (long)(wi * PH + py) * IMGW + (wj * PH + px);
#pragma unroll
    for (int vvv = 0; vvv < 8; ++vvv) {
      const int m0 = vvv + mhi, m1 = 16 + vvv + mhi;
      const float xv0 = x[(((long)(b * NC + m0)) << 16) + gpix];
      const float xv1 = x[(((long)(b * NC + m1)) << 16) + gpix];
      BUFB[m0 * NPIX + n] = xv0 + (d0[vvv] + B3V[m0]) * BTA[m0];
      BUFB[m1 * NPIX + n] = xv1 + (d1[vvv] + B3V[m1]) * BTA[m1];
    }
  }

  // --- LN2 over channels, per pixel column (wave-local) -> BUFA ---
  for (int i = 0; i < 4; ++i) {
    const int p = pbase + i * 32 + lane;
    float s1 = 0.f, s2 = 0.f;
    for (int c = 0; c < 32; ++c) { float v = BUFB[c * NPIX + p]; s1 += v; s2 += v * v; }
    const float mu = s1 * (1.f / 32.f);
    const float rs = rsqrtf(s2 * (1.f / 32.f) - mu * mu + 1e-6f);
    for (int c = 0; c < 32; ++c) {
      float v = BUFB[c * NPIX + p];
      BUFA[c * NPIX + p] = (v - mu) * rs * LG2[c] + LB2[c];
    }
  }

  // --- GEMM2 (w4) + fragment-level gate + GEMM3 (w5) + residual out ---
  for (int nt = wvid * 8; nt < wvid * 8 + 8; ++nt) {
    const int n = nt * 16 + l15;
    v8f e0 = {0.f,0.f,0.f,0.f,0.f,0.f,0.f,0.f};
    v8f e1 = e0, e2 = e0, e3 = e0;
#pragma unroll
    for (int kc = 0; kc < 8; ++kc) {
      const int c0 = kc * 4 + khalf;
      v2f bb; bb.x = BUFA[c0 * NPIX + n]; bb.y = BUFA[(c0 + 1) * NPIX + n];
      v2f a0, a1, a2, a3;
      a0.x = W4N[l15 * 32 + c0];        a0.y = W4N[l15 * 32 + c0 + 1];
      a1.x = W4N[(16 + l15) * 32 + c0]; a1.y = W4N[(16 + l15) * 32 + c0 + 1];
      a2.x = W4N[(32 + l15) * 32 + c0]; a2.y = W4N[(32 + l15) * 32 + c0 + 1];
      a3.x = W4N[(48 + l15) * 32 + c0]; a3.y = W4N[(48 + l15) * 32 + c0 + 1];
      e0 = wmma4(a0, bb, e0); e1 = wmma4(a1, bb, e1);
      e2 = wmma4(a2, bb, e2); e3 = wmma4(a3, bb, e3);
    }
    // gate: ch k pairs with k+32 -> same lane & VGPR slot of (e0,e2)/(e1,e3)
#pragma unroll
    for (int vvv = 0; vvv < 8; ++vvv) {
      const int m = vvv + mhi;
      BUFA[m * NPIX + n]        = (e0[vvv] + B4V[m])      * (e2[vvv] + B4V[32 + m]);
      BUFA[(16 + m) * NPIX + n] = (e1[vvv] + B4V[16 + m]) * (e3[vvv] + B4V[48 + m]);
    }
    // GEMM3 on gated h (same columns, same wave -> in-order LDS)
    v8f f0 = {0.f,0.f,0.f,0.f,0.f,0.f,0.f,0.f};
    v8f f1 = f0;
#pragma unroll
    for (int kc = 0; kc < 8; ++kc) {
      const int c0 = kc * 4 + khalf;
      v2f bb; bb.x = BUFA[c0 * NPIX + n]; bb.y = BUFA[(c0 + 1) * NPIX + n];
      v2f a0, a1;
      a0.x = W5N[l15 * 32 + c0];        a0.y = W5N[l15 * 32 + c0 + 1];
      a1.x = W5N[(16 + l15) * 32 + c0]; a1.y = W5N[(16 + l15) * 32 + c0 + 1];
      f0 = wmma4(a0, bb, f0); f1 = wmma4(a1, bb, f1);
    }
    const int py = n >> 5, px = n & 31;
    const long gpix = (long)(wi * PH + py) * IMGW + (wj * PH + px);
#pragma unroll
    for (int vvv = 0; vvv < 8; ++vvv) {
      const int m0 = vvv + mhi, m1 = 16 + vvv + mhi;
      outp[(((long)(b * NC + m0)) << 16) + gpix] =
          BUFB[m0 * NPIX + n] + (f0[vvv] + B5V[m0]) * GMA[m0];
      outp[(((long)(b * NC + m1)) << 16) + gpix] =
          BUFB[m1 * NPIX + n] + (f1[vvv] + B5V[m1]) * GMA[m1];
    }
  }
}

// ---------------------------------------------------------------------------
extern "C" void kernel_launch(void* const* d_in, const int* in_sizes, int n_in,
                              void* d_out, int out_size, void* d_ws, size_t ws_size,
                              hipStream_t stream) {
  (void)in_sizes; (void)n_in; (void)out_size; (void)ws_size;
  const float* x    = (const float*)d_in[0];
  const float* mod1 = (const float*)d_in[1];
  const float* mod2 = (const float*)d_in[2];
  const float* mod3 = (const float*)d_in[3];
  const float* mod4 = (const float*)d_in[4];
  const float* mod5 = (const float*)d_in[5];
  const float* w1   = (const float*)d_in[6];
  const float* b1   = (const float*)d_in[7];
  const float* w2   = (const float*)d_in[8];
  const float* b2   = (const float*)d_in[9];
  const float* w3   = (const float*)d_in[10];
  const float* b3   = (const float*)d_in[11];
  const float* w4   = (const float*)d_in[12];
  const float* b4   = (const float*)d_in[13];
  const float* w5   = (const float*)d_in[14];
  const float* b5   = (const float*)d_in[15];
  const float* ln1g = (const float*)d_in[16];
  const float* ln1b = (const float*)d_in[17];
  const float* ln2g = (const float*)d_in[18];
  const float* ln2b = (const float*)d_in[19];
  const float* scaw = (const float*)d_in[20];
  const float* scab = (const float*)d_in[21];
  const float* beta = (const float*)d_in[22];
  const float* gamma= (const float*)d_in[23];
  float* outp = (float*)d_out;

  float* y1buf = (float*)d_ws;                          // 8*32*65536 floats
  float* pp    = y1buf + (size_t)NB * NC * IMG2;        // 8*32*64
  float* attb  = pp + (size_t)NB * NC * 64;             // 8*32

  const size_t lds1 = (size_t)(2 * NC * RPIX + 1024 + 1024 + 384 + 4 * 32 + 256) * sizeof(float);
  const size_t lds3 = (size_t)(2 * NC * NPIX + 1024 + 2048 + 1024 + 32 + 64 + 32 + 5 * 32) * sizeof(float);

  k1_ln_conv1_dw3_gate<<<NB * 64, 256, lds1, stream>>>(
      x, mod1, mod2, w1, b1, w2, b2, ln1g, ln1b, y1buf, pp);
  k2_sca<<<1, 256, 0, stream>>>(pp, scaw, scab, attb);
  k3_sca_apply_ffn<<<NB * 64, 256, lds3, stream>>>(
      x, y1buf, attb, mod3, mod4, mod5, w3, b3, w4, b4, w5, b5,
      ln2g, ln2b, beta, gamma, outp);
}